// DeeperGCN_87978110091845
// MI455X (gfx1250) — compile-verified
//
#include <hip/hip_runtime.h>
#include <math.h>

#define Nn   50000
#define Ee   800000
#define INF_ 128
#define Hh   96
#define H2   192
#define Cc   47
#define Ll   4
#define EPS_MSG 1e-7f
#define LN_EPS  1e-5f
#define MPW  8   // M-tiles per wave in the GEMM

typedef __attribute__((ext_vector_type(2))) float v2f;
typedef __attribute__((ext_vector_type(8))) float v8f;

// ---------------- WMMA f32 GEMM, weight-stationary ----------------
// C[M x ldc] (+)= A[M x K] @ W[K x WNc] + bias
// Each wave holds ALL B fragments for its 16-wide N column in registers
// (K/4 x v2f), then streams MPW M-tiles: per K-step just one v2f A load
// and one v_wmma_f32_16x16x4_f32.
// grid: (ceil(WNc/16), ceil(ceil(mTiles/MPW)/8)), block: 256 (8 waves)
template <int K>
__global__ __launch_bounds__(256)
void wmma_gemm_f32(const float* __restrict__ A, const float* __restrict__ W,
                   const float* __restrict__ bias, float* __restrict__ C,
                   int WNc, int ldc, int ncValid, int accumulate, int mTiles)
{
    const int lane   = threadIdx.x & 31;
    const int tileN  = blockIdx.x;
    const int waveId = blockIdx.y * (blockDim.x >> 5) + (threadIdx.x >> 5);
    const int tm0    = waveId * MPW;
    if (tm0 >= mTiles) return;

    const int khalf = (lane >> 4) * 2;            // K sub-offset 0 or 2
    int colB = tileN * 16 + (lane & 15);
    if (colB > WNc - 1) colB = WNc - 1;           // clamp (store is guarded)

    // load all B fragments once (weights are tiny and L2-resident)
    v2f bfrag[K / 4];
#pragma unroll
    for (int kk = 0; kk < K / 4; ++kk) {
        bfrag[kk].x = W[(size_t)(kk * 4 + khalf)     * WNc + colB];
        bfrag[kk].y = W[(size_t)(kk * 4 + khalf + 1) * WNc + colB];
    }

    const int   cc = tileN * 16 + (lane & 15);
    const float bv = (bias != nullptr && cc < ncValid) ? bias[cc] : 0.0f;

    const int tmEnd = (tm0 + MPW < mTiles) ? (tm0 + MPW) : mTiles;
    for (int tm = tm0; tm < tmEnd; ++tm) {
        const float* __restrict__ Arow =
            A + (size_t)(tm * 16 + (lane & 15)) * K + khalf;
        v8f acc = {};
#pragma unroll
        for (int kk = 0; kk < K / 4; ++kk) {
            v2f a = *(const v2f*)(Arow + kk * 4);
            acc = __builtin_amdgcn_wmma_f32_16x16x4_f32(
                      false, a, false, bfrag[kk], (short)0, acc, false, false);
        }
        if (cc < ncValid) {
            const int rbase = tm * 16 + (lane >> 4) * 8;
#pragma unroll
            for (int r = 0; r < 8; ++r) {
                const size_t idx = (size_t)(rbase + r) * ldc + cc;
                const float  v   = acc[r] + bv;
                if (accumulate) C[idx] += v; else C[idx] = v;
            }
        }
    }
}

// ---------------- segment init ----------------
__global__ __launch_bounds__(256)
void init_seg(float* __restrict__ mx, float* __restrict__ den,
              float* __restrict__ num, int total)
{
    int i = blockIdx.x * blockDim.x + threadIdx.x;
    if (i < total) {
        mx[i]  = -__builtin_huge_valf();
        den[i] = 0.0f;
        num[i] = 0.0f;
    }
}

__device__ inline void atomicMaxF(float* addr, float val)
{
    if (val >= 0.0f) atomicMax((int*)addr, __float_as_int(val));
    else             atomicMin((unsigned int*)addr, __float_as_uint(val));
}

// ---------------- pass 1: per-(edge,4ch) segment max ----------------
__global__ __launch_bounds__(256)
void edge_max(const float* __restrict__ x, const int* __restrict__ src,
              const int* __restrict__ dst, const float* __restrict__ conv_t,
              int layer, float* __restrict__ mx)
{
    const int tid = blockIdx.x * blockDim.x + threadIdx.x;
    const int e  = tid / (Hh / 4);
    if (e >= Ee) return;
    const int cg = (tid % (Hh / 4)) * 4;
    const int s = src[e], d = dst[e];
    const float t = conv_t[layer];
    const float4 xv = *(const float4*)(x + (size_t)s * Hh + cg);
    float* mrow = mx + (size_t)d * Hh + cg;
    atomicMaxF(mrow + 0, (fmaxf(xv.x, 0.0f) + EPS_MSG) * t);
    atomicMaxF(mrow + 1, (fmaxf(xv.y, 0.0f) + EPS_MSG) * t);
    atomicMaxF(mrow + 2, (fmaxf(xv.z, 0.0f) + EPS_MSG) * t);
    atomicMaxF(mrow + 3, (fmaxf(xv.w, 0.0f) + EPS_MSG) * t);
}

// ---------------- pass 2: exp sums ----------------
__global__ __launch_bounds__(256)
void edge_expsum(const float* __restrict__ x, const int* __restrict__ src,
                 const int* __restrict__ dst, const float* __restrict__ conv_t,
                 int layer, const float* __restrict__ mx,
                 float* __restrict__ den, float* __restrict__ num)
{
    const int tid = blockIdx.x * blockDim.x + threadIdx.x;
    const int e  = tid / (Hh / 4);
    if (e >= Ee) return;
    const int cg = (tid % (Hh / 4)) * 4;
    const int s = src[e], d = dst[e];
    const float t = conv_t[layer];
    const float4 xv = *(const float4*)(x + (size_t)s * Hh + cg);
    const float4 mv = *(const float4*)(mx + (size_t)d * Hh + cg);
    float* drow = den + (size_t)d * Hh + cg;
    float* nrow = num + (size_t)d * Hh + cg;
#pragma unroll
    for (int i = 0; i < 4; ++i) {
        const float xi  = (i == 0) ? xv.x : (i == 1) ? xv.y : (i == 2) ? xv.z : xv.w;
        const float mxi = (i == 0) ? mv.x : (i == 1) ? mv.y : (i == 2) ? mv.z : mv.w;
        const float msg = fmaxf(xi, 0.0f) + EPS_MSG;
        const float ex  = __expf(msg * t - mxi);
        atomicAdd(drow + i, ex);
        atomicAdd(nrow + i, ex * msg);
    }
}

// ---------------- combine: out = num/den + x ----------------
__global__ __launch_bounds__(256)
void seg_combine(const float* __restrict__ x, const float* __restrict__ num,
                 const float* __restrict__ den, float* __restrict__ out, int total)
{
    int i = blockIdx.x * blockDim.x + threadIdx.x;
    if (i < total) {
        const float d = den[i];
        out[i] = num[i] / (d > 0.0f ? d : 1.0f) + x[i];
    }
}

// ---------------- LayerNorm (+optional ReLU), one wave per row ----------------
__global__ __launch_bounds__(256)
void ln_relu(const float* __restrict__ in, float* __restrict__ out,
             const float* __restrict__ g, const float* __restrict__ b,
             int rows, int width, int doRelu)
{
    const int wavesPerBlock = blockDim.x >> 5;
    const int row  = blockIdx.x * wavesPerBlock + (threadIdx.x >> 5);
    const int lane = threadIdx.x & 31;
    if (row >= rows) return;
    const float* __restrict__ r = in + (size_t)row * width;
    float s = 0.0f, sq = 0.0f;
    for (int c = lane; c < width; c += 32) {
        const float v = r[c];
        s += v; sq += v * v;
    }
#pragma unroll
    for (int off = 16; off > 0; off >>= 1) {
        s  += __shfl_xor(s,  off, 32);
        sq += __shfl_xor(sq, off, 32);
    }
    const float mu  = s / (float)width;
    const float var = sq / (float)width - mu * mu;
    const float inv = rsqrtf(var + LN_EPS);
    float* __restrict__ o = out + (size_t)row * width;
    for (int c = lane; c < width; c += 32) {
        float v = (r[c] - mu) * inv * g[c] + b[c];
        if (doRelu) v = fmaxf(v, 0.0f);
        o[c] = v;
    }
}

extern "C" void kernel_launch(void* const* d_in, const int* in_sizes, int n_in,
                              void* d_out, int out_size, void* d_ws, size_t ws_size,
                              hipStream_t stream)
{
    const float* features = (const float*)d_in[0];
    const int*   edge     = (const int*)  d_in[1];   // [2, E] (src row then dst row)
    const float* enc_w    = (const float*)d_in[2];
    const float* enc_b    = (const float*)d_in[3];
    const float* conv_t   = (const float*)d_in[4];
    const float* mlp_w1   = (const float*)d_in[5];   // [L, 96, 192]
    const float* mlp_b1   = (const float*)d_in[6];   // [L, 192]
    const float* mlp_ln_g = (const float*)d_in[7];
    const float* mlp_ln_b = (const float*)d_in[8];
    const float* mlp_w2   = (const float*)d_in[9];   // [L, 192, 96]
    const float* mlp_b2   = (const float*)d_in[10];  // [L, 96]
    const float* norm_g   = (const float*)d_in[11];  // [L, 96]
    const float* norm_b   = (const float*)d_in[12];
    const float* lin_w    = (const float*)d_in[13];  // [96, 47]
    const float* lin_b    = (const float*)d_in[14];
    float* out = (float*)d_out;

    const int* src = edge;
    const int* dst = edge + Ee;

    // workspace carve-up (all f32, ~134 MB)
    float* p   = (float*)d_ws;
    float* h   = p; p += (size_t)Nn * Hh;   // running node state
    float* x   = p; p += (size_t)Nn * Hh;   // conv input (post pre-norm)
    float* mxb = p; p += (size_t)Nn * Hh;   // segment max, then reused as conv-out
    float* den = p; p += (size_t)Nn * Hh;
    float* num = p; p += (size_t)Nn * Hh;
    float* z1  = p; p += (size_t)Nn * H2;   // MLP hidden

    const dim3 blk256(256);
    const int mTiles     = Nn / 16;                             // 3125
    const int gemmY      = ((mTiles + MPW - 1) / MPW + 7) / 8;  // waves->blocks
    const int nodeElems  = Nn * Hh;
    const int nodeBlocks = (nodeElems + 255) / 256;
    const int edgeBlocks = (Ee * (Hh / 4) + 255) / 256;
    const int lnBlocks   = (Nn + 7) / 8;                        // 8 waves/block

    // encoder: h = features @ enc_w + enc_b
    wmma_gemm_f32<INF_><<<dim3(Hh / 16, gemmY), blk256, 0, stream>>>(
        features, enc_w, enc_b, h, Hh, Hh, Hh, 0, mTiles);

    for (int layer = 0; layer < Ll; ++layer) {
        const float* xin;
        if (layer == 0) {
            xin = h;                                      // raw conv, no pre-norm
        } else {
            ln_relu<<<lnBlocks, blk256, 0, stream>>>(
                h, x, norm_g + layer * Hh, norm_b + layer * Hh, Nn, Hh, 1);
            xin = x;
        }

        init_seg<<<nodeBlocks, blk256, 0, stream>>>(mxb, den, num, nodeElems);
        edge_max<<<edgeBlocks, blk256, 0, stream>>>(xin, src, dst, conv_t, layer, mxb);
        edge_expsum<<<edgeBlocks, blk256, 0, stream>>>(xin, src, dst, conv_t, layer,
                                                       mxb, den, num);
        // conv-out = agg + x   (written into mxb, which is dead now)
        seg_combine<<<nodeBlocks, blk256, 0, stream>>>(xin, num, den, mxb, nodeElems);

        // z1 = out @ w1 + b1 ; z1 = relu(LN(z1)) ; h (+)= z1 @ w2 + b2
        wmma_gemm_f32<Hh><<<dim3(H2 / 16, gemmY), blk256, 0, stream>>>(
            mxb, mlp_w1 + (size_t)layer * Hh * H2, mlp_b1 + layer * H2,
            z1, H2, H2, H2, 0, mTiles);
        ln_relu<<<lnBlocks, blk256, 0, stream>>>(
            z1, z1, mlp_ln_g + layer * H2, mlp_ln_b + layer * H2, Nn, H2, 1);
        wmma_gemm_f32<H2><<<dim3(Hh / 16, gemmY), blk256, 0, stream>>>(
            z1, mlp_w2 + (size_t)layer * H2 * Hh, mlp_b2 + layer * Hh,
            h, Hh, Hh, Hh, layer == 0 ? 0 : 1, mTiles);
    }

    // final: relu(LN(h, norm[0])) @ lin_w + lin_b
    ln_relu<<<lnBlocks, blk256, 0, stream>>>(h, x, norm_g, norm_b, Nn, Hh, 1);
    wmma_gemm_f32<Hh><<<dim3((Cc + 15) / 16, gemmY), blk256, 0, stream>>>(
        x, lin_w, lin_b, out, Cc, Cc, Cc, 0, mTiles);
}